// MotifEnhancer_39127152066814
// MI455X (gfx1250) — compile-verified
//
#include <hip/hip_runtime.h>
#include <hip/hip_bf16.h>
#include <math.h>

typedef float v2f __attribute__((ext_vector_type(2)));
typedef float v8f __attribute__((ext_vector_type(8)));

#define IN_C   128
#define OUT_C  128
#define HEADS  4
#define HC     (HEADS * OUT_C)        // 512 projected channels
#define LDS_STRIDE 132                // 128 + 4 pad -> bank-conflict-free A reads

// ---------------------------------------------------------------------------
// atomic float max via CAS (emax initialized to -inf)
__device__ __forceinline__ void atomicMaxF(float* addr, float val) {
    unsigned int* ua = (unsigned int*)addr;
    unsigned int old = *ua;
    while (__uint_as_float(old) < val) {
        unsigned int assumed = old;
        old = atomicCAS(ua, assumed, __float_as_uint(val));
        if (old == assumed) break;
    }
}

// ---------------------------------------------------------------------------
// init: zero d_out, emax = -inf, denom = 0
__global__ void init_kernel(float* __restrict__ out, int n_out,
                            float* __restrict__ emax, float* __restrict__ denom, int n_ed) {
    int stride = gridDim.x * blockDim.x;
    for (int i = blockIdx.x * blockDim.x + threadIdx.x; i < n_out; i += stride)
        out[i] = 0.0f;
    for (int i = blockIdx.x * blockDim.x + threadIdx.x; i < n_ed; i += stride) {
        emax[i]  = -INFINITY;
        denom[i] = 0.0f;
    }
}

// ---------------------------------------------------------------------------
// V[i,h] = sum_c W[i, h*OUT_C + c] * att[h, c]   (folds att into a [128,4] matrix)
// t in [0,512): V_src; t in [512,1024): V_dst.  V layout: V[i*4 + h].
__global__ void prep_v_kernel(const float* __restrict__ Wsrc, const float* __restrict__ Wdst,
                              const float* __restrict__ att_s, const float* __restrict__ att_d,
                              float* __restrict__ V) {
    int t = blockIdx.x * blockDim.x + threadIdx.x;
    if (t >= 2 * IN_C * HEADS) return;
    const float* W   = (t < IN_C * HEADS) ? Wsrc  : Wdst;
    const float* att = (t < IN_C * HEADS) ? att_s : att_d;
    int u = t & (IN_C * HEADS - 1);
    int i = u >> 2, h = u & 3;
    float acc = 0.0f;
    #pragma unroll 4
    for (int c = 0; c < OUT_C; ++c)
        acc += W[(size_t)i * HC + h * OUT_C + c] * att[h * OUT_C + c];
    V[t] = acc;
}

// ---------------------------------------------------------------------------
// a[n,h] = sum_i X[n,i] * V[i,h]   — one wave (32 lanes) per row, float4 loads
__launch_bounds__(256)
__global__ void proj_a_kernel(const float* __restrict__ X, const float* __restrict__ V,
                              float* __restrict__ a, int N) {
    int row  = (blockIdx.x * blockDim.x + threadIdx.x) >> 5;
    int lane = threadIdx.x & 31;
    if (row >= N) return;
    float4 x = ((const float4*)X)[(size_t)row * 32 + lane];
    const float4* V4 = (const float4*)V;            // V4[i] = {h0,h1,h2,h3}
    float xs[4] = {x.x, x.y, x.z, x.w};
    float a0 = 0.f, a1 = 0.f, a2 = 0.f, a3 = 0.f;
    #pragma unroll
    for (int j = 0; j < 4; ++j) {
        float4 vv = V4[lane * 4 + j];
        a0 += xs[j] * vv.x;  a1 += xs[j] * vv.y;
        a2 += xs[j] * vv.z;  a3 += xs[j] * vv.w;
    }
    #pragma unroll
    for (int off = 16; off > 0; off >>= 1) {
        a0 += __shfl_xor(a0, off, 32);
        a1 += __shfl_xor(a1, off, 32);
        a2 += __shfl_xor(a2, off, 32);
        a3 += __shfl_xor(a3, off, 32);
    }
    if (lane == 0) ((float4*)a)[row] = make_float4(a0, a1, a2, a3);
}

// ---------------------------------------------------------------------------
// H[N,512] = X[N,128] @ W[128,512] using V_WMMA_F32_16X16X4_F32.
// Block = 256 threads = 8 waves; block tile = 32 rows x 512 cols.
// Wave w: row strip (w&1)*16, col group (w>>1)*128 -> 8 16x16 C tiles (64 VGPRs).
__launch_bounds__(256)
__global__ void gemm_h_kernel(const float* __restrict__ X, const float* __restrict__ W,
                              float* __restrict__ H, int N) {
    __shared__ float sX[32 * LDS_STRIDE];

    const int tid  = threadIdx.x;
    const int wave = tid >> 5;
    const int lane = tid & 31;
    const int row0 = blockIdx.x * 32;

    // stage X tile [32 x 128] into padded LDS (float4 cooperative load)
    for (int i = tid; i < 32 * 32; i += 256) {
        int r = i >> 5, q = i & 31;
        int gr = row0 + r;
        float4 v = make_float4(0.f, 0.f, 0.f, 0.f);
        if (gr < N) v = ((const float4*)X)[(size_t)gr * 32 + q];
        *(float4*)&sX[r * LDS_STRIDE + q * 4] = v;
    }
    __syncthreads();

    const int rt   = wave & 1;                 // row tile within block
    const int cg   = wave >> 1;                // 128-col group
    const int col0 = cg * 128;
    const int lrow = rt * 16 + (lane & 15);    // A-frag row (M = lane%16)
    const int kof  = (lane >> 4) * 2;          // K sub-offset: 0 (lanes 0-15) / 2 (16-31)
    const int bn   = lane & 15;                // B-frag col (N = lane%16)

    v8f acc[8] = {};

    for (int k = 0; k < IN_C; k += 4) {
        // A fragment: A[M, k+kof], A[M, k+kof+1] -> ds_load_b64 (bank-conflict-free)
        v2f afrag = *(const v2f*)&sX[lrow * LDS_STRIDE + k + kof];
        #pragma unroll
        for (int t = 0; t < 8; ++t) {
            // B fragment: W[k+kof+{0,1}, col0 + t*16 + bn]
            const float* wp = W + (size_t)(k + kof) * HC + col0 + t * 16 + bn;
            v2f bfrag;
            bfrag.x = wp[0];
            bfrag.y = wp[HC];
            acc[t] = __builtin_amdgcn_wmma_f32_16x16x4_f32(
                false, afrag, false, bfrag, (short)0, acc[t], false, false);
        }
    }

    // store C/D: lane l, vgpr v -> row = v + 8*(l>=16), col = t*16 + (l&15)
    const int rbase = row0 + rt * 16 + ((lane >> 4) * 8);
    #pragma unroll
    for (int t = 0; t < 8; ++t) {
        #pragma unroll
        for (int v = 0; v < 8; ++v) {
            int gr = rbase + v;
            if (gr < N) H[(size_t)gr * HC + col0 + t * 16 + bn] = acc[t][v];
        }
    }
}

// ---------------------------------------------------------------------------
// per-edge logits + segment max:  e = lrelu_0.01(a_src[s]+a_dst[d]); atomicMax emax[d]
__global__ void edge_max_kernel(const int* __restrict__ idx,
                                const float* __restrict__ a_src, const float* __restrict__ a_dst,
                                float* __restrict__ emax, int E) {
    int e = blockIdx.x * blockDim.x + threadIdx.x;
    if (e >= E) return;
    int s = idx[e], d = idx[E + e];
    float4 as = ((const float4*)a_src)[s];
    float4 ad = ((const float4*)a_dst)[d];
    float l[4] = {as.x + ad.x, as.y + ad.y, as.z + ad.z, as.w + ad.w};
    #pragma unroll
    for (int h = 0; h < 4; ++h) {
        float v = l[h] > 0.f ? l[h] : 0.01f * l[h];
        atomicMaxF(&emax[(size_t)d * 4 + h], v);
    }
}

// ---------------------------------------------------------------------------
// ex = exp(e - emax[d]); store numerator; atomicAdd denom[d]
__global__ void edge_ex_kernel(const int* __restrict__ idx,
                               const float* __restrict__ a_src, const float* __restrict__ a_dst,
                               const float* __restrict__ emax,
                               float* __restrict__ exb, float* __restrict__ denom, int E) {
    int e = blockIdx.x * blockDim.x + threadIdx.x;
    if (e >= E) return;
    int s = idx[e], d = idx[E + e];
    float4 as = ((const float4*)a_src)[s];
    float4 ad = ((const float4*)a_dst)[d];
    float4 mx = ((const float4*)emax)[d];
    float l[4] = {as.x + ad.x, as.y + ad.y, as.z + ad.z, as.w + ad.w};
    float m[4] = {mx.x, mx.y, mx.z, mx.w};
    float ex[4];
    #pragma unroll
    for (int h = 0; h < 4; ++h) {
        float v = l[h] > 0.f ? l[h] : 0.01f * l[h];
        ex[h] = __expf(v - m[h]);
        atomicAdd(&denom[(size_t)d * 4 + h], ex[h]);
    }
    ((float4*)exb)[e] = make_float4(ex[0], ex[1], ex[2], ex[3]);
}

// ---------------------------------------------------------------------------
// message scatter with fused head-mean: one wave per edge, lane covers 4 channels.
// out[d,c] += 0.25 * sum_h alpha[h] * H[s, h*128 + c]
__launch_bounds__(256)
__global__ void scatter_kernel(const float* __restrict__ Hs, const int* __restrict__ idx,
                               const float* __restrict__ exb, const float* __restrict__ denom,
                               float* __restrict__ out, int E) {
    int e    = (blockIdx.x * blockDim.x + threadIdx.x) >> 5;
    int lane = threadIdx.x & 31;
    if (e >= E) return;
    int s = idx[e], d = idx[E + e];
    float4 ex = ((const float4*)exb)[e];
    float4 dn = ((const float4*)denom)[d];
    float al0 = ex.x / (dn.x + 1e-16f);
    float al1 = ex.y / (dn.y + 1e-16f);
    float al2 = ex.z / (dn.z + 1e-16f);
    float al3 = ex.w / (dn.w + 1e-16f);
    const float* hp = Hs + (size_t)s * HC;
    float* op = out + (size_t)d * OUT_C;
    #pragma unroll
    for (int j = 0; j < 4; ++j) {
        int c = lane + j * 32;
        float v = al0 * hp[c]       + al1 * hp[128 + c]
                + al2 * hp[256 + c] + al3 * hp[384 + c];
        atomicAdd(op + c, 0.25f * v);
    }
}

// ---------------------------------------------------------------------------
// out = lrelu_0.1(out + bias)
__global__ void finalize_kernel(float* __restrict__ out, const float* __restrict__ bias, int total) {
    int i = blockIdx.x * blockDim.x + threadIdx.x;
    if (i >= total) return;
    float v = out[i] + bias[i & (OUT_C - 1)];
    out[i] = v > 0.f ? v : 0.1f * v;
}

// ---------------------------------------------------------------------------
extern "C" void kernel_launch(void* const* d_in, const int* in_sizes, int n_in,
                              void* d_out, int out_size, void* d_ws, size_t ws_size,
                              hipStream_t stream) {
    const float* fra_vec = (const float*)d_in[0];
    const float* mol_vec = (const float*)d_in[1];
    const int*   idx     = (const int*)  d_in[2];
    const float* W_src   = (const float*)d_in[3];
    // d_in[4] = W_dst (only used via V_dst), d_in[5] = att_src, d_in[6] = att_dst
    const float* W_dst   = (const float*)d_in[4];
    const float* att_src = (const float*)d_in[5];
    const float* att_dst = (const float*)d_in[6];
    const float* bias    = (const float*)d_in[7];

    const int N_SRC = in_sizes[0] / IN_C;
    const int N_DST = in_sizes[1] / IN_C;
    const int E     = in_sizes[2] / 2;

    // workspace layout (256B aligned)
    size_t off = 0;
    auto wsAlloc = [&](size_t bytes) -> char* {
        char* p = (char*)d_ws + off;
        off = (off + bytes + 255) & ~(size_t)255;
        return p;
    };
    float* h_src = (float*)wsAlloc((size_t)N_SRC * HC * sizeof(float));   // ~205 MB
    float* a_src = (float*)wsAlloc((size_t)N_SRC * HEADS * sizeof(float));
    float* a_dst = (float*)wsAlloc((size_t)N_DST * HEADS * sizeof(float));
    float* emax  = (float*)wsAlloc((size_t)N_DST * HEADS * sizeof(float));
    float* denom = (float*)wsAlloc((size_t)N_DST * HEADS * sizeof(float));
    float* exb   = (float*)wsAlloc((size_t)E * HEADS * sizeof(float));
    float* Vmat  = (float*)wsAlloc((size_t)2 * IN_C * HEADS * sizeof(float)); // V_src | V_dst
    (void)ws_size;

    float* out = (float*)d_out;

    // 1. init out / emax / denom
    init_kernel<<<1024, 256, 0, stream>>>(out, out_size, emax, denom, N_DST * HEADS);

    // 2. fold att into V matrices [128,4]
    prep_v_kernel<<<4, 256, 0, stream>>>(W_src, W_dst, att_src, att_dst, Vmat);

    // 3. per-node attention scalars a = X @ V  (wave per row)
    {
        int blocks_s = (int)(((size_t)N_SRC * 32 + 255) / 256);
        int blocks_d = (int)(((size_t)N_DST * 32 + 255) / 256);
        proj_a_kernel<<<blocks_s, 256, 0, stream>>>(fra_vec, Vmat,           a_src, N_SRC);
        proj_a_kernel<<<blocks_d, 256, 0, stream>>>(mol_vec, Vmat + IN_C*HEADS, a_dst, N_DST);
    }

    // 4. WMMA fp32 GEMM: h_src = fra_vec @ W_src
    gemm_h_kernel<<<(N_SRC + 31) / 32, 256, 0, stream>>>(fra_vec, W_src, h_src, N_SRC);

    // 5. segment max of leaky logits
    edge_max_kernel<<<(E + 255) / 256, 256, 0, stream>>>(idx, a_src, a_dst, emax, E);

    // 6. softmax numerators + denominators
    edge_ex_kernel<<<(E + 255) / 256, 256, 0, stream>>>(idx, a_src, a_dst, emax, exb, denom, E);

    // 7. weighted message scatter with fused head-mean (wave per edge)
    scatter_kernel<<<(int)(((size_t)E * 32 + 255) / 256), 256, 0, stream>>>(
        h_src, idx, exb, denom, out, E);

    // 8. bias + module LeakyReLU(0.1)
    finalize_kernel<<<(out_size + 255) / 256, 256, 0, stream>>>(out, bias, out_size);
}